// TrainableTree_79611513798985
// MI455X (gfx1250) — compile-verified
//
#include <hip/hip_runtime.h>
#include <hip/hip_bf16.h>

typedef __attribute__((ext_vector_type(2))) float v2f;
typedef __attribute__((ext_vector_type(8))) float v8f;

// ---------------------------------------------------------------------------
// Phase 1: collapse the linear tree into w_eff (64 floats) + c_eff (1 float).
// Dual-vector propagation from the root down:
//   v      = Wr row                      (dual of L1 outputs)
//   u1[j]  = W1[j]^T v                   (dual of L2 outputs, j=0..1)
//   u2[k]  = W2[k]^T u1[k/2]             (dual of L3 outputs, k=0..3)
//   u3[k]  = W3[k]^T u2[k/2]             (dual of leaf outputs, k=0..7)
//   w_eff  = Σ_k Wleaf[k]^T u3[k/2]      (applied to sin(h))
//   c_eff  = br + Σ b1·v + Σ b2·u1 + Σ b3·u2 + Σ bleaf·u3
// ---------------------------------------------------------------------------
__global__ void tree_collapse_kernel(
    const float* __restrict__ Wleaf, const float* __restrict__ bleaf,
    const float* __restrict__ W3,    const float* __restrict__ b3,
    const float* __restrict__ W2,    const float* __restrict__ b2,
    const float* __restrict__ W1,    const float* __restrict__ b1,
    const float* __restrict__ Wr,    const float* __restrict__ br,
    float* __restrict__ ws)           // ws[0..63] = w_eff, ws[64] = c_eff
{
    __shared__ float v[64];
    __shared__ float u1[2][64];
    __shared__ float u2[4][64];
    __shared__ float u3[8][64];
    __shared__ float weff[64];
    __shared__ float cacc[256];

    const int tid = threadIdx.x;
    cacc[tid] = 0.0f;
    if (tid < 64) v[tid] = Wr[tid];
    __syncthreads();

    // level-1 duals (2 x 64 outputs) + b1 bias dot
    if (tid < 128) {
        const int j = tid >> 6, o = tid & 63;
        float s = 0.0f;
        for (int p = 0; p < 64; ++p) s += W1[j * 4096 + p * 64 + o] * v[p];
        u1[j][o] = s;
        cacc[tid] += b1[tid] * v[o];
    }
    __syncthreads();

    // level-2 duals (4 x 64 outputs) + b2 bias dot
    {
        const int k = tid >> 6, o = tid & 63;
        const float* uu = u1[k >> 1];
        float s = 0.0f;
        for (int p = 0; p < 64; ++p) s += W2[k * 4096 + p * 64 + o] * uu[p];
        u2[k][o] = s;
        cacc[tid] += b2[tid] * uu[o];
    }
    __syncthreads();

    // level-3 duals (8 x 64 outputs, two passes) + b3 bias dot
    for (int half = 0; half < 2; ++half) {
        const int idx = half * 256 + tid;
        const int k = idx >> 6, o = idx & 63;
        const float* uu = u2[k >> 1];
        float s = 0.0f;
        for (int p = 0; p < 64; ++p) s += W3[k * 4096 + p * 64 + o] * uu[p];
        u3[k][o] = s;
        cacc[tid] += b3[idx] * uu[o];
    }
    __syncthreads();

    // leaves: w_eff[m] = Σ_k Σ_p Wleaf[k][p,m] * u3[k/2][p]
    if (tid < 64) {
        float s = 0.0f;
        for (int k = 0; k < 16; ++k) {
            const float* uu = u3[k >> 1];
            const float* W  = Wleaf + k * 4096;
            for (int p = 0; p < 64; ++p) s += W[p * 64 + tid] * uu[p];
        }
        weff[tid] = s;
    }
    // bleaf bias dots (16 x 64 terms, four passes)
    for (int q = 0; q < 4; ++q) {
        const int idx = q * 256 + tid;
        const int k = idx >> 6, o = idx & 63;
        cacc[tid] += bleaf[idx] * u3[k >> 1][o];
    }
    __syncthreads();

    if (tid == 0) {
        float c = br[0];
        for (int i = 0; i < 256; ++i) c += cacc[i];
        ws[64] = c;
    }
    if (tid < 64) ws[tid] = weff[tid];
}

// ---------------------------------------------------------------------------
// Phase 2: streaming pass. Per 16-sample slab (one wave):
//   h = x @ Wt^T + bt  via V_WMMA_F32_16X16X4_F32 (K: 10 padded to 12, 3 steps,
//       4 output-column tiles of 16)
//   res = Σ_o sin(h[:,o]) * w_eff[o]   (in-register + shfl_xor reduction)
//   out = a_envelope * (res + c_eff)
// 256 threads = 8 wave32 waves; 128 samples per block.
// ---------------------------------------------------------------------------
__global__ void tree_main_kernel(
    const float* __restrict__ x,  const float* __restrict__ Wt,
    const float* __restrict__ bt, const float* __restrict__ wc,
    float* __restrict__ out, int N)
{
    __shared__ float wtT[12][64];     // wtT[k][o] = Wt[o,k], rows 10..11 zero
    __shared__ float btS[64];
    __shared__ float weffS[64];
    __shared__ float xs[8][16][12];   // per-wave x tile, zero-padded K

    const int tid  = threadIdx.x;
    const int wave = tid >> 5;
    const int lane = tid & 31;
    const int n0   = blockIdx.x * 128 + wave * 16;

    for (int i = tid; i < 12 * 64; i += 256) {
        const int k = i >> 6, o = i & 63;
        wtT[k][o] = (k < 10) ? Wt[o * 10 + k] : 0.0f;
    }
    if (tid < 64) { btS[tid] = bt[tid]; weffS[tid] = wc[tid]; }

    for (int i = lane; i < 16 * 12; i += 32) {
        const int si = i / 12, k = i % 12;
        const int n = n0 + si;
        xs[wave][si][k] = (k < 10 && n < N) ? x[n * 10 + k] : 0.0f;
    }
    __syncthreads();

    const int m  = lane & 15;   // A-row / B-column index within tile
    const int hi = lane >> 4;   // lane half selects K pair (f32 WMMA layout)

    v8f acc[4] = {};
    for (int s = 0; s < 3; ++s) {
        const int kb = 4 * s + 2 * hi;       // K = kb + r  (r = VGPR index 0..1)
        v2f a;
        a.x = xs[wave][m][kb];
        a.y = xs[wave][m][kb + 1];
        for (int t = 0; t < 4; ++t) {
            v2f b;
            b.x = wtT[kb][16 * t + m];
            b.y = wtT[kb + 1][16 * t + m];
            acc[t] = __builtin_amdgcn_wmma_f32_16x16x4_f32(
                false, a, false, b, (short)0, acc[t], false, false);
        }
    }

    // D layout: acc[t][r] = h[sample = r + 8*hi][o = 16*t + m]
    float p[8];
    for (int r = 0; r < 8; ++r) p[r] = 0.0f;
    for (int t = 0; t < 4; ++t) {
        const int o = 16 * t + m;
        const float w  = weffS[o];
        const float bb = btS[o];
        for (int r = 0; r < 8; ++r) {
            const float h = acc[t][r] + bb;
            p[r] += __sinf(h) * w;
        }
    }
    // reduce over the 16 lanes of each half (each half covers all 64 o's)
    for (int off = 1; off < 16; off <<= 1)
        for (int r = 0; r < 8; ++r)
            p[r] += __shfl_xor(p[r], off, 32);

    // gather: lane l (0..15) wants sample l; lanes<16 hold samples 0..7 (r),
    // lanes>=16 hold samples 8..15 (r)
    float resv = 0.0f;
    for (int r = 0; r < 8; ++r) {
        const float v0 = __shfl(p[r], 0, 32);
        const float v1 = __shfl(p[r], 16, 32);
        if ((lane & 7) == r) resv = (lane & 8) ? v1 : v0;
    }

    const float ceff = wc[64];
    const int n = n0 + lane;
    if (lane < 16 && n < N) {
        float prod = 1.0f;
        for (int k = 0; k < 10; ++k) {
            const float xv = xs[wave][lane][k];
            prod *= (xv * xv - 1.0f);
        }
        const float aenv = prod * rsqrtf(1000.0f + prod * prod);
        out[n] = aenv * (resv + ceff);
    }
}

extern "C" void kernel_launch(void* const* d_in, const int* in_sizes, int n_in,
                              void* d_out, int out_size, void* d_ws, size_t ws_size,
                              hipStream_t stream) {
    const float* x     = (const float*)d_in[0];
    const float* Wt    = (const float*)d_in[1];
    const float* bt    = (const float*)d_in[2];
    const float* Wleaf = (const float*)d_in[3];
    const float* bleaf = (const float*)d_in[4];
    const float* W3    = (const float*)d_in[5];
    const float* b3    = (const float*)d_in[6];
    const float* W2    = (const float*)d_in[7];
    const float* b2    = (const float*)d_in[8];
    const float* W1    = (const float*)d_in[9];
    const float* b1    = (const float*)d_in[10];
    const float* Wr    = (const float*)d_in[11];
    const float* br    = (const float*)d_in[12];
    float* out = (float*)d_out;
    float* ws  = (float*)d_ws;

    const int N = in_sizes[0] / 10;

    tree_collapse_kernel<<<1, 256, 0, stream>>>(Wleaf, bleaf, W3, b3, W2, b2,
                                                W1, b1, Wr, br, ws);
    const int grid = (N + 127) / 128;
    tree_main_kernel<<<grid, 256, 0, stream>>>(x, Wt, bt, ws, out, N);
}